// QLSTM_65481071402744
// MI455X (gfx1250) — compile-verified
//
#include <hip/hip_runtime.h>
#include <hip/hip_bf16.h>

typedef float v2f __attribute__((ext_vector_type(2)));
typedef float v8f __attribute__((ext_vector_type(8)));

#define T_TOT 1024
#define B_TOT 4096
#define D_IN  16
#define H_DIM 4
#define NG    16   // 4 gates * 4 wires

// CDNA5 hardware tanh (TRANS op). Per ISA §7.4: 1 independent op / delay after
// a TRANS op before its result is consumed -> emit s_delay_alu with the asm.
__device__ __forceinline__ float tanh_hw(float x) {
  float r;
  asm volatile("v_tanh_f32 %0, %1\n\ts_delay_alu instid0(TRANS32_DEP_1)"
               : "=v"(r) : "v"(x));
  return r;
}

// ---------------------------------------------------------------------------
// Phase 1: pre[row,16] = X[row,16] @ Wx^T + (b + rx), rows = tc*B, via
// V_WMMA_F32_16X16X4_F32 (16x16 M/N tile, K=4 x 4 k-tiles, exact fp32).
// A layout (f32 16x4): lane<16 -> M=lane, K={0,1}; lane>=16 -> M=lane-16, K={2,3}
// B layout (f32 4x16): lane<16 -> N=lane, K={0,1}; lane>=16 -> N=lane-16, K={2,3}
// C/D layout: vgpr r: lanes0-15 (M=r,N=lane), lanes16-31 (M=r+8,N=lane-16)
// ---------------------------------------------------------------------------
__global__ void qlstm_gemm_wmma(const float* __restrict__ X,
                                float* __restrict__ pre,
                                const float* __restrict__ Wf, const float* __restrict__ Wi,
                                const float* __restrict__ Wg, const float* __restrict__ Wo,
                                const float* __restrict__ bf, const float* __restrict__ bi,
                                const float* __restrict__ bg, const float* __restrict__ bo,
                                const float* __restrict__ rxf, const float* __restrict__ rxi,
                                const float* __restrict__ rxg, const float* __restrict__ rxo,
                                long ntiles)
{
  const int lane = threadIdx.x & 31;
  const int n    = lane & 15;          // M index for A, N index for B/C/D
  const int kh   = (lane >> 4) << 1;   // 0 or 2: which K-pair this half-wave owns
  const int moff = (lane >> 4) << 3;   // 0 or 8: D-row offset

  // Per-lane weight column (B operand) and bias, hoisted out of the tile loop.
  const int gate = n >> 2, wire = n & 3;
  const float* W  = (gate == 0) ? Wf : (gate == 1) ? Wi : (gate == 2) ? Wg : Wo;
  const float* bb = (gate == 0) ? bf : (gate == 1) ? bi : (gate == 2) ? bg : bo;
  const float* rx = (gate == 0) ? rxf : (gate == 1) ? rxi : (gate == 2) ? rxg : rxo;
  const float bias = bb[wire] + rx[wire];

  v2f bmat[4];
#pragma unroll
  for (int kt = 0; kt < 4; ++kt) {
    const int k0 = kt * 4 + kh;        // k < 16: x-part of the [4,20] weight rows
    bmat[kt].x = W[wire * 20 + k0];
    bmat[kt].y = W[wire * 20 + k0 + 1];
  }

  const long wid = (long)blockIdx.x * (blockDim.x >> 5) + (threadIdx.x >> 5);
  const long nw  = (long)gridDim.x * (blockDim.x >> 5);

  for (long tile = wid; tile < ntiles; tile += nw) {
    const long rowbase = tile << 4;
    const float* xr = X + (rowbase + n) * D_IN;   // this lane's A-matrix row

    v8f c;
#pragma unroll
    for (int r = 0; r < 8; ++r) c[r] = bias;      // C = broadcast bias_n

#pragma unroll
    for (int kt = 0; kt < 4; ++kt) {
      const float2 af = *reinterpret_cast<const float2*>(xr + kt * 4 + kh);
      v2f a; a.x = af.x; a.y = af.y;
      c = __builtin_amdgcn_wmma_f32_16x16x4_f32(false, a, false, bmat[kt],
                                                (short)0, c, false, false);
    }

    float* pr = pre + rowbase * NG;
#pragma unroll
    for (int r = 0; r < 8; ++r)
      pr[(size_t)(r + moff) * NG + n] = c[r];     // half-wave writes one 64B row
  }
}

// ---------------------------------------------------------------------------
// Phase 2: sequential LSTM recurrence. 4 lanes per batch element (one gate
// each, 8 elements/wave) to minimize the serial per-step instruction count.
// theta = pre + hx@Wh ; z = cumprod(cos(theta)) ; act via hw tanh ; shfl share.
// ---------------------------------------------------------------------------
__global__ void qlstm_recur(const float* __restrict__ pre,
                            float* __restrict__ out,
                            float* __restrict__ hxs, float* __restrict__ cxs,
                            float* __restrict__ hx_fin, float* __restrict__ cx_fin,
                            const float* __restrict__ Wf, const float* __restrict__ Wi,
                            const float* __restrict__ Wg, const float* __restrict__ Wo,
                            int t0, int tc, int first)
{
  const int g    = threadIdx.x & 3;                    // gate: 0=f 1=i 2=g 3=o
  const int b    = blockIdx.x * 64 + (threadIdx.x >> 2);
  const int lane = threadIdx.x & 31;
  const int base = lane & ~3;                          // first lane of this quad

  const float* W = (g == 0) ? Wf : (g == 1) ? Wi : (g == 2) ? Wg : Wo;
  float wh[4][4];                                      // hx part: cols 16..19
#pragma unroll
  for (int w = 0; w < 4; ++w)
#pragma unroll
    for (int j = 0; j < 4; ++j)
      wh[w][j] = W[w * 20 + D_IN + j];

  // sigmoid(x) = 0.5 + 0.5*tanh(0.5x); gate 'g' (tanh) uses (1,1,0).
  const float si  = (g == 2) ? 1.0f : 0.5f;
  const float so  = si;
  const float off = (g == 2) ? 0.0f : 0.5f;

  float hx[4], cx[4];
  if (first) {
#pragma unroll
    for (int w = 0; w < 4; ++w) { hx[w] = 0.0f; cx[w] = 0.0f; }
  } else {
#pragma unroll
    for (int w = 0; w < 4; ++w) { hx[w] = hxs[b * 4 + w]; cx[w] = cxs[b * 4 + w]; }
  }

  const float* pp = pre + (size_t)b * NG + g * 4;      // per-step stride B*NG
  float4 pv = *reinterpret_cast<const float4*>(pp);    // software pipeline: t=0

  for (int t = 0; t < tc; ++t) {
    float4 nv = pv;
    if (t + 1 < tc)                                    // prefetch next step
      nv = *reinterpret_cast<const float4*>(pp + (size_t)(t + 1) * B_TOT * NG);

    float th[4] = {pv.x, pv.y, pv.z, pv.w};
#pragma unroll
    for (int w = 0; w < 4; ++w)
#pragma unroll
      for (int j = 0; j < 4; ++j)
        th[w] = fmaf(hx[j], wh[w][j], th[w]);

    // <Z_w> = prod_{j<=w} cos(theta_j)   (CNOT chain closed form)
    float c0 = __cosf(th[0]), c1 = __cosf(th[1]);
    float c2 = __cosf(th[2]), c3 = __cosf(th[3]);
    float a[4];
    a[0] = off + so * tanh_hw(si * c0);
    float z1 = c0 * c1;       a[1] = off + so * tanh_hw(si * z1);
    float z2 = z1 * c2;       a[2] = off + so * tanh_hw(si * z2);
    float z3 = z2 * c3;       a[3] = off + so * tanh_hw(si * z3);

    // gather all four gates' activations for this element (quad of lanes)
    float F[4], I[4], G[4], O[4];
#pragma unroll
    for (int w = 0; w < 4; ++w) {
      F[w] = __shfl(a[w], base + 0, 32);
      I[w] = __shfl(a[w], base + 1, 32);
      G[w] = __shfl(a[w], base + 2, 32);
      O[w] = __shfl(a[w], base + 3, 32);
    }
#pragma unroll
    for (int w = 0; w < 4; ++w) {
      cx[w] = fmaf(F[w], cx[w], I[w] * G[w]);
      hx[w] = O[w] * tanh_hw(cx[w]);
    }

    if (g == 0) {
      float4 o4 = make_float4(hx[0], hx[1], hx[2], hx[3]);
      *reinterpret_cast<float4*>(out + ((size_t)(t0 + t) * B_TOT + b) * H_DIM) = o4;
    }
    pv = nv;
  }

  if (g == 0) {
#pragma unroll
    for (int w = 0; w < 4; ++w) {
      hxs[b * 4 + w] = hx[w];  cxs[b * 4 + w] = cx[w];       // chunk carry
      hx_fin[b * 4 + w] = hx[w];  cx_fin[b * 4 + w] = cx[w]; // final outputs
    }
  }
}

extern "C" void kernel_launch(void* const* d_in, const int* in_sizes, int n_in,
                              void* d_out, int out_size, void* d_ws, size_t ws_size,
                              hipStream_t stream) {
  (void)in_sizes; (void)n_in; (void)out_size;
  const float* X   = (const float*)d_in[0];
  const float* Wf  = (const float*)d_in[1];
  const float* bfv = (const float*)d_in[2];
  const float* Wi  = (const float*)d_in[3];
  const float* biv = (const float*)d_in[4];
  const float* Wg  = (const float*)d_in[5];
  const float* bgv = (const float*)d_in[6];
  const float* Wo  = (const float*)d_in[7];
  const float* bov = (const float*)d_in[8];
  const float* rxf = (const float*)d_in[9];
  const float* rxi = (const float*)d_in[10];
  const float* rxg = (const float*)d_in[11];
  const float* rxo = (const float*)d_in[12];
  float* out = (float*)d_out;

  float* hxs = (float*)d_ws;                 // [B,4] carry between chunks
  float* cxs = hxs + (size_t)B_TOT * H_DIM;
  float* pre = cxs + (size_t)B_TOT * H_DIM;  // [Tc, B, 16] staging

  const size_t state_b = (size_t)2 * B_TOT * H_DIM * sizeof(float);
  const size_t per_t   = (size_t)B_TOT * NG * sizeof(float);
  long cap = (ws_size > state_b) ? (long)((ws_size - state_b) / per_t) : 0;
  int Tc = (int)(cap < 1 ? 1 : (cap > T_TOT ? T_TOT : cap));
  if (Tc > 128) Tc = 128;  // 32MB chunk: keep pre resident in 192MB L2

  float* hx_fin = out + (size_t)T_TOT * B_TOT * H_DIM;
  float* cx_fin = hx_fin + (size_t)B_TOT * H_DIM;

  int t0 = 0, first = 1;
  while (t0 < T_TOT) {
    const int tc = (T_TOT - t0 < Tc) ? (T_TOT - t0) : Tc;
    const long ntiles = (long)tc * B_TOT / 16;
    int blocks = (int)((ntiles + 7) / 8);
    if (blocks > 4096) blocks = 4096;
    qlstm_gemm_wmma<<<blocks, 256, 0, stream>>>(
        X + (size_t)t0 * B_TOT * D_IN, pre,
        Wf, Wi, Wg, Wo, bfv, biv, bgv, bov, rxf, rxi, rxg, rxo, ntiles);
    qlstm_recur<<<B_TOT / 64, 256, 0, stream>>>(
        pre, out, hxs, cxs, hx_fin, cx_fin, Wf, Wi, Wg, Wo, t0, tc, first);
    first = 0;
    t0 += tc;
  }
}